// SGSNetLoss_42013370089717
// MI455X (gfx1250) — compile-verified
//
#include <hip/hip_runtime.h>
#include <hip/hip_bf16.h>

// ---------------------------------------------------------------------------
// YOLO-style loss for MI455X (gfx1250).
// Dense part reads ONLY the 3 objectness channels (0, 85, 170) of each tensor
// (2.7 MB instead of 232 MB). Sparse part gathers bbox/cls logits at the 8
// positive cells per batch. All reductions are fixed-order (deterministic).
// Wave-level reduction uses V_WMMA_F32_16X16X4_F32 (full fp32 precision).
// ---------------------------------------------------------------------------

#define BATCH   64
#define NTGT    8
#define NCLS    80

typedef __attribute__((ext_vector_type(2))) float v2f;
typedef __attribute__((ext_vector_type(8))) float v8f;

__device__ __forceinline__ float softplusf(float x) {
    // stable: logaddexp(x, 0)
    return fmaxf(x, 0.0f) + log1pf(expf(-fabsf(x)));
}

// Wave32 sum via V_WMMA_F32_16X16X4_F32 with A = all-ones:
//   D[m,n] = sum_k B[k,n]  ->  sum_n D[0,n] = sum of all 64 B entries
//                             = sum of per-lane partials (other half is 0).
// C/D layout: VGPR0 holds M=0 (lanes 0-15, N=lane) and M=8 (lanes 16-31),
// and D rows are identical, so reducing d[0] over each 16-lane half gives the
// wave total on lanes 0 and 16.
__device__ __forceinline__ float wave_sum(float partial) {
#if defined(__gfx1250__) && __has_builtin(__builtin_amdgcn_wmma_f32_16x16x4_f32)
    v2f a; a[0] = 1.0f;    a[1] = 1.0f;
    v2f b; b[0] = partial; b[1] = 0.0f;
    v8f c = {};
    v8f d = __builtin_amdgcn_wmma_f32_16x16x4_f32(
        /*neg_a=*/false, a, /*neg_b=*/false, b,
        /*c_mod=*/(short)0, c, /*reuse_a=*/false, /*reuse_b=*/false);
    float colsum = d[0];
    for (int off = 8; off > 0; off >>= 1)
        colsum += __shfl_down(colsum, off, 16);
    return colsum;                    // valid on lane 0 (and lane 16)
#else
    for (int off = 16; off > 0; off >>= 1)
        partial += __shfl_down(partial, off, 32);
    return partial;
#endif
}

// ---------------------------------------------------------------------------
// Kernel 1: dense objectness softplus sums.
// grid = (192, 3): blockIdx.y = scale, blockIdx.x = batch*3 + anchor.
// Each block reduces one H*W obj plane (channel a*85 of batch b).
// ---------------------------------------------------------------------------
__global__ void dense_obj_kernel(const float* __restrict__ p3,
                                 const float* __restrict__ p4,
                                 const float* __restrict__ p5,
                                 float* __restrict__ wsDense) {
    const int s  = blockIdx.y;
    const int H  = (s == 0) ? 52 : (s == 1) ? 26 : 13;
    const int HW = H * H;
    const float* p = (s == 0) ? p3 : (s == 1) ? p4 : p5;

    const int b = blockIdx.x / 3;
    const int a = blockIdx.x % 3;
    const float* plane = p + (size_t)(b * 255 + a * 85) * (size_t)HW;

    float partial = 0.0f;
    for (int i = threadIdx.x; i < HW; i += blockDim.x)
        partial += softplusf(plane[i]);

    // EXEC reconverges here; WMMA executed by the full block (8 waves).
    float wsum = wave_sum(partial);

    __shared__ float sh[8];
    const int lane = threadIdx.x & 31;
    const int wave = threadIdx.x >> 5;
    if (lane == 0) sh[wave] = wsum;
    __syncthreads();
    if (threadIdx.x == 0) {
        float t = 0.0f;
        for (int w = 0; w < 8; ++w) t += sh[w];   // fixed order
        wsDense[s * 192 + blockIdx.x] = t;
    }
}

// ---------------------------------------------------------------------------
// Kernel 2: positive-cell terms. grid = (64, 3): blockIdx.x = batch,
// blockIdx.y = scale. 128 threads/block.
// Emits per (scale,batch): [sum of positive obj logits,
//                           per_b_bbox (already / max(4*cnt,1)),
//                           per_b_cls  (already / max(80*cnt,1))]
// ---------------------------------------------------------------------------
__global__ void pos_kernel(const float* __restrict__ p3,
                           const float* __restrict__ p4,
                           const float* __restrict__ p5,
                           const float* __restrict__ boxes,
                           const int*   __restrict__ labels,
                           float* __restrict__ wsPos) {
    const int b  = blockIdx.x;
    const int s  = blockIdx.y;
    const int H  = (s == 0) ? 52 : (s == 1) ? 26 : 13;
    const int W  = H;
    const int HW = H * W;
    const float* p    = (s == 0) ? p3 : (s == 1) ? p4 : p5;
    const float* base = p + (size_t)b * 255u * (size_t)HW;

    // Every thread computes target metadata (cheap, avoids broadcasts).
    int   cell[NTGT];
    float tv[NTGT][4];
    bool  win[NTGT];
    for (int n = 0; n < NTGT; ++n) {
        const float* bx = boxes + ((size_t)b * NTGT + n) * 4;
        const float cx = bx[0], cy = bx[1], w = bx[2], h = bx[3];
        int gx = (int)floorf(cx * (float)W);
        int gy = (int)floorf(cy * (float)H);
        gx = min(max(gx, 0), W - 1);
        gy = min(max(gy, 0), H - 1);
        cell[n]  = gy * W + gx;
        tv[n][0] = cx * (float)W - (float)gx;
        tv[n][1] = cy * (float)H - (float)gy;
        tv[n][2] = w;
        tv[n][3] = h;
    }
    // ".at[].set" semantics: last writer to a cell wins; cnt = #distinct cells.
    int cnt = 0;
    for (int n = 0; n < NTGT; ++n) {
        bool wn = true;
        for (int m = n + 1; m < NTGT; ++m)
            if (cell[m] == cell[n]) { wn = false; break; }
        win[n] = wn;
        cnt += wn ? 1 : 0;
    }

    const int tid = threadIdx.x;          // 128 threads
    float accPos = 0.0f, accBox = 0.0f, accCls = 0.0f;

    // positive obj logits (channel 0, anchor 0)
    for (int n = tid; n < NTGT; n += blockDim.x)
        if (win[n]) accPos += base[cell[n]];

    // bbox squared error (channels 1..4, anchor 0)
    for (int i = tid; i < NTGT * 4; i += blockDim.x) {
        const int n = i >> 2, k = i & 3;
        if (win[n]) {
            const float x = base[(size_t)(1 + k) * HW + cell[n]];
            const float d = x - tv[n][k];
            accBox += d * d;
        }
    }

    // class BCE (channels 5..84, anchor 0)
    for (int i = tid; i < NTGT * NCLS; i += blockDim.x) {
        const int n = i / NCLS, c = i % NCLS;
        if (win[n]) {
            const float x = base[(size_t)(5 + c) * HW + cell[n]];
            const float t = (c == labels[b * NTGT + n]) ? 1.0f : 0.0f;
            accCls += softplusf(x) - t * x;
        }
    }

    __shared__ float sA[128], sB[128], sC[128];
    sA[tid] = accPos; sB[tid] = accBox; sC[tid] = accCls;
    __syncthreads();
    if (tid == 0) {
        float P = 0.0f, Bx = 0.0f, Cl = 0.0f;
        for (int i = 0; i < 128; ++i) { P += sA[i]; Bx += sB[i]; Cl += sC[i]; }
        const float fcnt = (float)cnt;
        float* e = wsPos + (size_t)(s * BATCH + b) * 3;
        e[0] = P;
        e[1] = Bx / fmaxf(fcnt * 4.0f, 1.0f);
        e[2] = Cl / fmaxf(fcnt * (float)NCLS, 1.0f);
    }
}

// ---------------------------------------------------------------------------
// Kernel 3: fold all partials (fixed order) into the 4 scalar outputs.
// ---------------------------------------------------------------------------
__global__ void final_kernel(const float* __restrict__ wsDense,
                             const float* __restrict__ wsPos,
                             float* __restrict__ out) {
    if (threadIdx.x != 0 || blockIdx.x != 0) return;
    float obj = 0.0f, bbox = 0.0f, cls = 0.0f;
    for (int s = 0; s < 3; ++s) {
        const int H = (s == 0) ? 52 : (s == 1) ? 26 : 13;
        float dsum = 0.0f;
        for (int j = 0; j < 192; ++j) dsum += wsDense[s * 192 + j];
        float px = 0.0f, bb = 0.0f, cl = 0.0f;
        for (int b = 0; b < BATCH; ++b) {
            const float* e = wsPos + (size_t)(s * BATCH + b) * 3;
            px += e[0]; bb += e[1]; cl += e[2];
        }
        obj  += (dsum - px) / ((float)BATCH * 3.0f * (float)(H * H));
        bbox += bb / (float)(BATCH * NTGT);
        cls  += cl / (float)(BATCH * NTGT);
    }
    obj /= 3.0f; bbox /= 3.0f; cls /= 3.0f;
    out[0] = obj + 5.0f * bbox + 2.0f * cls;
    out[1] = obj;
    out[2] = bbox;
    out[3] = cls;
}

// ---------------------------------------------------------------------------
extern "C" void kernel_launch(void* const* d_in, const int* in_sizes, int n_in,
                              void* d_out, int out_size, void* d_ws, size_t ws_size,
                              hipStream_t stream) {
    (void)in_sizes; (void)n_in; (void)out_size; (void)ws_size;
    const float* p3     = (const float*)d_in[0];
    const float* p4     = (const float*)d_in[1];
    const float* p5     = (const float*)d_in[2];
    const float* boxes  = (const float*)d_in[3];
    const int*   labels = (const int*)d_in[4];
    float* out = (float*)d_out;

    float* ws      = (float*)d_ws;
    float* wsDense = ws;          // 3 * 192 floats
    float* wsPos   = ws + 576;    // 3 * 64 * 3 floats

    dense_obj_kernel<<<dim3(192, 3), 256, 0, stream>>>(p3, p4, p5, wsDense);
    pos_kernel     <<<dim3(BATCH, 3), 128, 0, stream>>>(p3, p4, p5, boxes, labels, wsPos);
    final_kernel   <<<1, 32, 0, stream>>>(wsDense, wsPos, out);
}